// BertLabelAttentionCRFNER_62328565399970
// MI455X (gfx1250) — compile-verified
//
#include <hip/hip_runtime.h>
#include <math.h>

// ---------------- problem constants ----------------
constexpr int B_ = 8, T_ = 1024, L_ = 4, I_ = 4, H_ = 768;
constexpr int K_ = 3, HEADS_ = 4, DHEAD_ = H_ / HEADS_;
constexpr int N_ = T_ + L_ + I_;                 // 1032 nodes / sample
constexpr int TK_ = T_ * K_;                     // 3072
constexpr int EDYN_ = 4 * TK_;                   // 12288 dynamic edges
constexpr int EST_ = 2 * (T_ - 1) + 2 * I_ * L_ + I_ * (I_ - 1) + L_ * (L_ - 1) + N_; // 3134
constexpr int E_ = EDYN_ + EST_;                 // 15422 edges / sample
constexpr float NEG_SLOPE_ = 0.2f;

typedef float v2f __attribute__((ext_vector_type(2)));
typedef float v8f __attribute__((ext_vector_type(8)));

// ---------------- helpers ----------------
__device__ __forceinline__ void atomicMaxF(float* addr, float v) {
  // works for mixed signs: positive floats ordered as ints, negative as inverted uints
  if (v >= 0.0f) atomicMax((int*)addr, __float_as_int(v));
  else           atomicMin((unsigned int*)addr, __float_as_uint(v));
}

// ---------------- 0. pack x = concat(text, label, image) ----------------
__global__ void pack_x(const float* __restrict__ text, const float* __restrict__ lab,
                       const float* __restrict__ img, float* __restrict__ X) {
  long long i = (long long)blockIdx.x * blockDim.x + threadIdx.x;
  if (i >= (long long)B_ * N_ * H_) return;
  int c = (int)(i % H_);
  long long rem = i / H_;
  int n = (int)(rem % N_);
  int b = (int)(rem / N_);
  float v;
  if (n < T_)            v = text[((size_t)b * T_ + n) * H_ + c];
  else if (n < T_ + L_)  v = lab[((size_t)b * L_ + (n - T_)) * H_ + c];
  else                   v = img[((size_t)b * I_ + (n - T_ - L_)) * H_ + c];
  X[i] = v;
}

// ---------------- 1. dynamic edges: cosine top-k (one wave / text row) ----------------
__global__ void build_dyn_edges(const float* __restrict__ text, const float* __restrict__ lab,
                                const float* __restrict__ img,
                                int* __restrict__ esrc, int* __restrict__ edst) {
  const int t = blockIdx.x, b = blockIdx.y, lane = threadIdx.x;
  const float* tr = text + ((size_t)b * T_ + t) * H_;
  float tn = 0.f;
  float dl[4] = {0, 0, 0, 0}, nl[4] = {0, 0, 0, 0};
  float di[4] = {0, 0, 0, 0}, ni[4] = {0, 0, 0, 0};
  for (int c = lane; c < H_; c += 32) {
    float tv = tr[c];
    tn += tv * tv;
#pragma unroll
    for (int j = 0; j < 4; ++j) {
      float lv = lab[((size_t)b * L_ + j) * H_ + c];
      float iv = img[((size_t)b * I_ + j) * H_ + c];
      dl[j] += tv * lv; nl[j] += lv * lv;
      di[j] += tv * iv; ni[j] += iv * iv;
    }
  }
#pragma unroll
  for (int off = 16; off; off >>= 1) {
    tn += __shfl_xor(tn, off, 32);
#pragma unroll
    for (int j = 0; j < 4; ++j) {
      dl[j] += __shfl_xor(dl[j], off, 32);
      nl[j] += __shfl_xor(nl[j], off, 32);
      di[j] += __shfl_xor(di[j], off, 32);
      ni[j] += __shfl_xor(ni[j], off, 32);
    }
  }
  if (lane == 0) {
    float tnorm = sqrtf(tn);
    float cl[4], ci[4];
#pragma unroll
    for (int j = 0; j < 4; ++j) {
      cl[j] = dl[j] / fmaxf(tnorm * sqrtf(nl[j]), 1e-8f);
      ci[j] = di[j] / fmaxf(tnorm * sqrtf(ni[j]), 1e-8f);
    }
    int il[4] = {0, 1, 2, 3}, im[4] = {0, 1, 2, 3};
    // stable selection sort (descending, strict > keeps lower index first on ties)
#pragma unroll
    for (int a = 0; a < 3; ++a) {
      int bl = a, bi = a;
      for (int c = a + 1; c < 4; ++c) {
        if (cl[il[c]] > cl[il[bl]]) bl = c;
        if (ci[im[c]] > ci[im[bi]]) bi = c;
      }
      int tmp = il[a]; il[a] = il[bl]; il[bl] = tmp;
      tmp = im[a]; im[a] = im[bi]; im[bi] = tmp;
    }
    const size_t base = (size_t)b * E_;
#pragma unroll
    for (int k = 0; k < K_; ++k) {
      int el = T_ + il[k];
      int ei = T_ + L_ + im[k];
      size_t p = base + (size_t)t * K_ + k;
      esrc[p] = t;                 edst[p] = el;              // ti -> el
      esrc[p + TK_] = el;          edst[p + TK_] = t;         // el -> ti
      esrc[p + 2 * TK_] = t;       edst[p + 2 * TK_] = ei;    // ti -> ei
      esrc[p + 3 * TK_] = ei;      edst[p + 3 * TK_] = t;     // ei -> ti
    }
  }
}

// ---------------- 2. static edges ----------------
__global__ void build_static_edges(int* __restrict__ esrc, int* __restrict__ edst) {
  int i = blockIdx.x * blockDim.x + threadIdx.x;
  if (i >= EST_) return;
  int s, d, r = i;
  if (r < T_ - 1) { s = r; d = r + 1; }
  else if ((r -= (T_ - 1)) < T_ - 1) { s = r + 1; d = r; }
  else if ((r -= (T_ - 1)) < I_ * L_) { int ii = r / L_, jj = r % L_; s = T_ + L_ + ii; d = T_ + jj; }
  else if ((r -= I_ * L_) < I_ * L_) { int ii = r / L_, jj = r % L_; s = T_ + jj; d = T_ + L_ + ii; }
  else if ((r -= I_ * L_) < I_ * (I_ - 1)) {
    int p = r / (I_ - 1), rem = r % (I_ - 1); int q = rem + (rem >= p);
    s = T_ + L_ + p; d = T_ + L_ + q;
  } else if ((r -= I_ * (I_ - 1)) < L_ * (L_ - 1)) {
    int p = r / (L_ - 1), rem = r % (L_ - 1); int q = rem + (rem >= p);
    s = T_ + p; d = T_ + q;
  } else { r -= L_ * (L_ - 1); s = r; d = r; }  // self loops
#pragma unroll
  for (int b = 0; b < B_; ++b) {
    esrc[(size_t)b * E_ + EDYN_ + i] = s;
    edst[(size_t)b * E_ + EDYN_ + i] = d;
  }
}

// ---------------- 3. GEMM h = x @ W via V_WMMA_F32_16X16X4_F32 ----------------
// block = 192 threads = 6 waves; each wave owns a 16(M) x 64(N) strip (4 WMMA tiles).
// The 16-row A tile is shared by all 6 waves, so it is staged into LDS with
// double-buffered GLOBAL_LOAD_ASYNC_TO_LDS_B128 (ASYNCcnt) in K-chunks of 48:
// 16 rows x 48 cols x 4B = 3072B = 192 threads x one b128 each.
// LDS row stride padded to 52 floats -> the 16-lane ds_load_b64 fragment read
// covers all 64 banks conflict-free (52*r mod 64 = 4*(13r mod 16)).
constexpr int GEMM_WAVES = 6;
constexpr int KC_ = 48;                 // K-chunk staged per double-buffer slot
constexpr int APAD_ = 52;               // padded LDS row stride (floats)
__global__ __launch_bounds__(192) void gat_gemm_wmma(const float* __restrict__ X,
                                                     const float* __restrict__ Wm,
                                                     float* __restrict__ Y) {
  __shared__ float As[2][16 * APAD_];
  const int tid = threadIdx.x;
  const int wave = tid >> 5, lane = tid & 31;
  const int m0 = blockIdx.x * 16;
  const int n0 = (blockIdx.y * GEMM_WAVES + wave) * 64;
  const int b = blockIdx.z;
  const float* A = X + (size_t)b * N_ * H_;
  float* C = Y + (size_t)b * N_ * H_;

  // cooperative async A staging: thread tid covers row tid/12, cols (tid%12)*4..+3
  const int arow_s = tid / 12;          // 0..15
  const int acol_s = (tid % 12) * 4;    // 0,4,...,44
  const int grow = m0 + arow_s;
  const float* aga = A + (size_t)(grow < N_ ? grow : 0) * H_ + acol_s;
  const unsigned ldsA0 = (unsigned)(uintptr_t)(&As[0][arow_s * APAD_ + acol_s]);
  const unsigned ldsA1 = (unsigned)(uintptr_t)(&As[1][arow_s * APAD_ + acol_s]);

  // prefetch chunk 0 into buffer 0 (async DMA into LDS, tracked by ASYNCcnt)
  asm volatile("global_load_async_to_lds_b128 %0, %1, off"
               :: "v"(ldsA0), "v"(aga) : "memory");

  const int coll = lane & 15;
  const int krow = (lane >> 4) << 1;    // lanes 0-15: K {k,k+1}; lanes 16-31: {k+2,k+3}
  const float* afrag0 = &As[0][coll * APAD_ + krow];
  const float* afrag1 = &As[1][coll * APAD_ + krow];

  v8f acc0 = {}, acc1 = {}, acc2 = {}, acc3 = {};

  asm volatile("s_wait_asynccnt 0x0" ::: "memory");
  __syncthreads();

  constexpr int NCHUNK = H_ / KC_;      // 16
  for (int c = 0; c < NCHUNK; ++c) {
    // prefetch next chunk into the other buffer while computing this one
    if (c + 1 < NCHUNK) {
      const unsigned dstl = (c & 1) ? ldsA0 : ldsA1;
      const float* gnext = aga + (c + 1) * KC_;
      asm volatile("global_load_async_to_lds_b128 %0, %1, off"
                   :: "v"(dstl), "v"(gnext) : "memory");
    }
    const float* ab = (c & 1) ? afrag1 : afrag0;
    const int kbase = c * KC_;
#pragma unroll
    for (int kk = 0; kk < KC_; kk += 4) {
      v2f a = *(const v2f*)(ab + kk);   // ds_load_b64, bank-conflict-free
      const float* wk0 = Wm + (size_t)(kbase + kk + krow) * H_;
      const float* wk1 = wk0 + H_;
      v2f b0, b1, b2, b3;
      b0.x = wk0[n0 + coll];      b0.y = wk1[n0 + coll];
      b1.x = wk0[n0 + 16 + coll]; b1.y = wk1[n0 + 16 + coll];
      b2.x = wk0[n0 + 32 + coll]; b2.y = wk1[n0 + 32 + coll];
      b3.x = wk0[n0 + 48 + coll]; b3.y = wk1[n0 + 48 + coll];
      acc0 = __builtin_amdgcn_wmma_f32_16x16x4_f32(false, a, false, b0, (short)0, acc0, false, false);
      acc1 = __builtin_amdgcn_wmma_f32_16x16x4_f32(false, a, false, b1, (short)0, acc1, false, false);
      acc2 = __builtin_amdgcn_wmma_f32_16x16x4_f32(false, a, false, b2, (short)0, acc2, false, false);
      acc3 = __builtin_amdgcn_wmma_f32_16x16x4_f32(false, a, false, b3, (short)0, acc3, false, false);
    }
    // publish next buffer / retire reads of current one before it is overwritten
    asm volatile("s_wait_asynccnt 0x0" ::: "memory");
    __syncthreads();
  }

  // C/D 16x16 layout: VGPR v -> rows m0+v (lanes 0-15) and m0+8+v (lanes 16-31)
  const int rbase = m0 + ((lane >> 4) << 3);
  v8f accs[4] = {acc0, acc1, acc2, acc3};
#pragma unroll
  for (int j = 0; j < 4; ++j) {
    const int col = n0 + 16 * j + coll;
#pragma unroll
    for (int v = 0; v < 8; ++v) {
      const int rr = rbase + v;
      if (rr < N_) C[(size_t)rr * H_ + col] = accs[j][v];
    }
  }
}

// ---------------- 4. per-node attention coefficients (one wave / head) ----------------
__global__ __launch_bounds__(128) void attn_coef(const float* __restrict__ HL,
                                                 const float* __restrict__ asrc,
                                                 const float* __restrict__ adst,
                                                 float* __restrict__ es, float* __restrict__ ed) {
  const int flat = blockIdx.x;                       // b*N + node
  const int head = threadIdx.x >> 5, lane = threadIdx.x & 31;
  const float* h = HL + (size_t)flat * H_ + head * DHEAD_;
  float s1 = 0.f, s2 = 0.f;
  for (int i = lane; i < DHEAD_; i += 32) {
    float v = h[i];
    s1 += v * asrc[head * DHEAD_ + i];
    s2 += v * adst[head * DHEAD_ + i];
  }
#pragma unroll
  for (int off = 16; off; off >>= 1) {
    s1 += __shfl_xor(s1, off, 32);
    s2 += __shfl_xor(s2, off, 32);
  }
  if (lane == 0) {
    es[(size_t)flat * HEADS_ + head] = s1;
    ed[(size_t)flat * HEADS_ + head] = s2;
  }
}

// ---------------- 5. per-layer init ----------------
__global__ void layer_init(float* __restrict__ acc, float* __restrict__ m, float* __restrict__ ss) {
  long long i = (long long)blockIdx.x * blockDim.x + threadIdx.x;
  if (i < (long long)B_ * N_ * H_) acc[i] = 0.f;
  if (i < (long long)B_ * N_ * HEADS_) { ss[i] = 0.f; m[i] = -INFINITY; }
}

// ---------------- 6. edge softmax: max ----------------
__global__ void edge_max(const int* __restrict__ esrc, const int* __restrict__ edst,
                         const float* __restrict__ es, const float* __restrict__ ed,
                         float* __restrict__ m) {
  long long i = (long long)blockIdx.x * blockDim.x + threadIdx.x;
  if (i >= (long long)B_ * E_) return;
  const int b = (int)(i / E_);
  const int s = esrc[i], d = edst[i];
  const size_t sb = ((size_t)b * N_ + s) * HEADS_, db = ((size_t)b * N_ + d) * HEADS_;
#pragma unroll
  for (int h = 0; h < HEADS_; ++h) {
    float v = es[sb + h] + ed[db + h];
    v = v > 0.f ? v : NEG_SLOPE_ * v;
    atomicMaxF(&m[db + h], v);
  }
}

// ---------------- 7. edge softmax: exp + sum ----------------
__global__ void edge_exp(const int* __restrict__ esrc, const int* __restrict__ edst,
                         const float* __restrict__ es, const float* __restrict__ ed,
                         const float* __restrict__ m, float* __restrict__ ex,
                         float* __restrict__ ss) {
  long long i = (long long)blockIdx.x * blockDim.x + threadIdx.x;
  if (i >= (long long)B_ * E_) return;
  const int b = (int)(i / E_);
  const int s = esrc[i], d = edst[i];
  const size_t sb = ((size_t)b * N_ + s) * HEADS_, db = ((size_t)b * N_ + d) * HEADS_;
#pragma unroll
  for (int h = 0; h < HEADS_; ++h) {
    float v = es[sb + h] + ed[db + h];
    v = v > 0.f ? v : NEG_SLOPE_ * v;
    float w = expf(v - m[db + h]);
    ex[(size_t)i * HEADS_ + h] = w;
    atomicAdd(&ss[db + h], w);
  }
}

// ---------------- 8. edge aggregation (block / edge) ----------------
__global__ __launch_bounds__(256) void edge_agg(const int* __restrict__ esrc,
                                                const int* __restrict__ edst,
                                                const float* __restrict__ ex,
                                                const float* __restrict__ ss,
                                                const float* __restrict__ HL,
                                                float* __restrict__ acc) {
  const long long i = blockIdx.x;  // b*E + e
  const int b = (int)(i / E_);
  const int s = esrc[i], d = edst[i];
  const size_t db = ((size_t)b * N_ + d) * HEADS_;
  float alpha[HEADS_];
#pragma unroll
  for (int h = 0; h < HEADS_; ++h) alpha[h] = ex[(size_t)i * HEADS_ + h] / ss[db + h];
  const float* hs = HL + ((size_t)b * N_ + s) * H_;
  float* od = acc + ((size_t)b * N_ + d) * H_;
  for (int c = threadIdx.x; c < H_; c += 256)
    atomicAdd(&od[c], alpha[c / DHEAD_] * hs[c]);
}

// ---------------- 9. bias + relu + residual + layernorm (block / node) ----------------
__global__ __launch_bounds__(256) void gat_post(float* __restrict__ X, const float* __restrict__ acc,
                                                const float* __restrict__ lbias,
                                                const float* __restrict__ g,
                                                const float* __restrict__ bv) {
  const int flat = blockIdx.x;  // b*N + node
  const float* a = acc + (size_t)flat * H_;
  float* x = X + (size_t)flat * H_;
  __shared__ float red[256];
  float y[3];
  float s = 0.f;
#pragma unroll
  for (int i = 0; i < 3; ++i) {
    int c = threadIdx.x + 256 * i;
    float v = fmaxf(a[c] + lbias[c], 0.f) + x[c];
    y[i] = v;
    s += v;
  }
  red[threadIdx.x] = s; __syncthreads();
  for (int off = 128; off; off >>= 1) {
    if (threadIdx.x < off) red[threadIdx.x] += red[threadIdx.x + off];
    __syncthreads();
  }
  const float mu = red[0] / H_;
  __syncthreads();
  s = 0.f;
#pragma unroll
  for (int i = 0; i < 3; ++i) { float dd = y[i] - mu; s += dd * dd; }
  red[threadIdx.x] = s; __syncthreads();
  for (int off = 128; off; off >>= 1) {
    if (threadIdx.x < off) red[threadIdx.x] += red[threadIdx.x + off];
    __syncthreads();
  }
  const float rstd = rsqrtf(red[0] / H_ + 1e-5f);
#pragma unroll
  for (int i = 0; i < 3; ++i) {
    int c = threadIdx.x + 256 * i;
    x[c] = (y[i] - mu) * rstd * g[c] + bv[c];
  }
}

// ---------------- 10. extract text rows ----------------
__global__ void extract_text(const float* __restrict__ X, float* __restrict__ out) {
  long long i = (long long)blockIdx.x * blockDim.x + threadIdx.x;
  if (i >= (long long)B_ * T_ * H_) return;
  int c = (int)(i % H_);
  long long rem = i / H_;
  int t = (int)(rem % T_);
  int b = (int)(rem / T_);
  out[i] = X[((size_t)b * N_ + t) * H_ + c];
}

// ---------------- host side ----------------
extern "C" void kernel_launch(void* const* d_in, const int* in_sizes, int n_in,
                              void* d_out, int out_size, void* d_ws, size_t ws_size,
                              hipStream_t stream) {
  (void)in_sizes; (void)n_in; (void)out_size; (void)ws_size;
  const float* text = (const float*)d_in[0];   // [B,T,H]
  const float* lab  = (const float*)d_in[1];   // [B,L,H]
  const float* img  = (const float*)d_in[2];   // [B,I,H]
  const float* W    = (const float*)d_in[3];   // [3,H,H]
  const float* asrc = (const float*)d_in[4];   // [3,HEADS,DHEAD]
  const float* adst = (const float*)d_in[5];   // [3,HEADS,DHEAD]
  const float* bias = (const float*)d_in[6];   // [3,H]
  const float* lng  = (const float*)d_in[7];   // [3,H]
  const float* lnb  = (const float*)d_in[8];   // [3,H]
  float* out = (float*)d_out;                  // [B,T,H]

  char* w = (char*)d_ws;
  auto takeF = [&](size_t n) { float* p = (float*)w; w += n * sizeof(float); return p; };
  auto takeI = [&](size_t n) { int* p = (int*)w; w += n * sizeof(int); return p; };
  const size_t NH = (size_t)B_ * N_ * H_;
  float* X    = takeF(NH);
  float* HL   = takeF(NH);
  float* ACC  = takeF(NH);
  float* ES   = takeF((size_t)B_ * N_ * HEADS_);
  float* ED   = takeF((size_t)B_ * N_ * HEADS_);
  float* M    = takeF((size_t)B_ * N_ * HEADS_);
  float* SS   = takeF((size_t)B_ * N_ * HEADS_);
  float* EX   = takeF((size_t)B_ * E_ * HEADS_);
  int* ESRC   = takeI((size_t)B_ * E_);
  int* EDST   = takeI((size_t)B_ * E_);

  const int nhBlocks = (int)((NH + 255) / 256);
  pack_x<<<nhBlocks, 256, 0, stream>>>(text, lab, img, X);
  build_dyn_edges<<<dim3(T_, B_), 32, 0, stream>>>(text, lab, img, ESRC, EDST);
  build_static_edges<<<(EST_ + 255) / 256, 256, 0, stream>>>(ESRC, EDST);

  const int eBlocks = (int)(((long long)B_ * E_ + 255) / 256);
  for (int li = 0; li < 3; ++li) {
    gat_gemm_wmma<<<dim3((N_ + 15) / 16, H_ / (GEMM_WAVES * 64), B_), 192, 0, stream>>>(
        X, W + (size_t)li * H_ * H_, HL);
    attn_coef<<<B_ * N_, 128, 0, stream>>>(HL, asrc + (size_t)li * H_, adst + (size_t)li * H_, ES, ED);
    layer_init<<<nhBlocks, 256, 0, stream>>>(ACC, M, SS);
    edge_max<<<eBlocks, 256, 0, stream>>>(ESRC, EDST, ES, ED, M);
    edge_exp<<<eBlocks, 256, 0, stream>>>(ESRC, EDST, ES, ED, M, EX, SS);
    edge_agg<<<B_ * E_, 256, 0, stream>>>(ESRC, EDST, EX, SS, HL, ACC);
    gat_post<<<B_ * N_, 256, 0, stream>>>(X, ACC, bias + (size_t)li * H_,
                                          lng + (size_t)li * H_, lnb + (size_t)li * H_);
  }
  extract_text<<<(int)(((long long)B_ * T_ * H_ + 255) / 256), 256, 0, stream>>>(X, out);
}